// StateSpaceModel_74586402063279
// MI455X (gfx1250) — compile-verified
//
#include <hip/hip_runtime.h>
#include <hip/hip_bf16.h>

#define G_ 128
#define T_ 128
#define S_ 64
#define M_ 8

typedef float v2f __attribute__((ext_vector_type(2)));
typedef float v8f __attribute__((ext_vector_type(8)));

// ---------------------------------------------------------------------------
// 64x64 f32 matmul on the WMMA pipe:  D = A * op(B) [+ Q] [+ KR*K^T]
// A, B, D are 64x64 row-major in LDS. 8 waves, 2 output tiles per wave,
// 16 K-slabs of V_WMMA_F32_16X16X4_F32 per tile.
// Fragment layouts per CDNA5 ISA 7.12.2:
//   A 16x4 : lane 0-15 M=lane, K={0,1}; lane 16-31 M=lane-16, K={2,3}
//   B 4x16 : row (K) striped across VGPRs, lanes 0-15 N, halves K{0,1}/{2,3}
//   C/D    : lane 0-15 N, vgpr r -> M=r (low half) / M=r+8 (high half)
// ---------------------------------------------------------------------------
__device__ __forceinline__ void mm64(const float* A, const float* B, float* D,
                                     bool transB,
                                     const float* addQ,   // optional epilogue +Q
                                     const float* KRp,    // optional epilogue
                                     const float* Kp)     //   + KR * K^T
{
    const int lane = threadIdx.x & 31;
    const int wave = threadIdx.x >> 5;
    const int half = lane >> 4;     // 0 or 1
    const int l15  = lane & 15;

    for (int tt = wave; tt < 16; tt += 8) {
        const int ti = (tt >> 2) * 16;   // tile row base (M)
        const int tj = (tt & 3) * 16;    // tile col base (N)
        v8f acc = {};
        #pragma unroll
        for (int kb = 0; kb < 16; ++kb) {
            const int k0 = kb * 4 + half * 2;
            v2f a, b;
            a.x = A[(ti + l15) * 64 + k0];
            a.y = A[(ti + l15) * 64 + k0 + 1];
            if (!transB) {
                b.x = B[(k0    ) * 64 + (tj + l15)];
                b.y = B[(k0 + 1) * 64 + (tj + l15)];
            } else {            // op(B)[k][n] = B[n][k]
                b.x = B[(tj + l15) * 64 + k0];
                b.y = B[(tj + l15) * 64 + k0 + 1];
            }
            acc = __builtin_amdgcn_wmma_f32_16x16x4_f32(
                      false, a, false, b, (short)0, acc, false, false);
        }
        #pragma unroll
        for (int r = 0; r < 8; ++r) {
            const int row = ti + r + half * 8;
            const int col = tj + l15;
            float v = acc[r];
            if (addQ) v += addQ[row * 64 + col];
            if (KRp) {
                #pragma unroll
                for (int m = 0; m < 8; ++m)
                    v += KRp[row * 8 + m] * Kp[col * 8 + m];
            }
            D[row * 64 + col] = v;
        }
    }
}

// ---------------------------------------------------------------------------
// Prologue: Q = Lq Lq^T, P0 = Linit Linit^T, R = Lr Lr^T  into workspace.
// ws layout (floats): Q[4096] | P0[4096] | R[64]
// ---------------------------------------------------------------------------
__global__ __launch_bounds__(256) void kf_prep(const float* __restrict__ Lq,
                                               const float* __restrict__ Lr,
                                               const float* __restrict__ Linit,
                                               float* __restrict__ ws)
{
    float* Q  = ws;
    float* P0 = ws + 4096;
    float* R  = ws + 8192;
    for (int e = blockIdx.x * blockDim.x + threadIdx.x; e < 4096;
         e += gridDim.x * blockDim.x) {
        const int i = e >> 6, j = e & 63;
        float q = 0.f, p = 0.f;
        for (int k = 0; k < 64; ++k) {
            q += Lq[i * 64 + k] * Lq[j * 64 + k];
            p += Linit[i * 64 + k] * Linit[j * 64 + k];
        }
        Q[e] = q;
        P0[e] = p;
    }
    const int e = blockIdx.x * blockDim.x + threadIdx.x;
    if (e < 64) {
        const int i = e >> 3, j = e & 7;
        float r = 0.f;
        for (int k = 0; k < 8; ++k) r += Lr[i * 8 + k] * Lr[j * 8 + k];
        R[e] = r;
    }
}

// ---------------------------------------------------------------------------
// Main Kalman scan: one workgroup (8 wave32) per group g.
// ---------------------------------------------------------------------------
__global__ __launch_bounds__(256) void kf_main(const float* __restrict__ y,
                                               const float* __restrict__ F,
                                               const float* __restrict__ H,
                                               const float* __restrict__ Qg,
                                               const float* __restrict__ Rg,
                                               const float* __restrict__ P0g,
                                               const float* __restrict__ init_mean,
                                               float* __restrict__ out)
{
    extern __shared__ float smem[];
    float* Fs    = smem;               // 4096  F, row-major
    float* bufA  = Fs   + 4096;        // 4096  rotating 64x64
    float* bufB  = bufA + 4096;        // 4096
    float* bufC  = bufB + 4096;        // 4096
    float* Hs    = bufC + 4096;        // 512   H[m][s]
    float* Rs    = Hs   + 512;         // 64    R 8x8
    float* PHt   = Rs   + 64;          // 512   cov_p @ H^T   [s][m]
    float* Km    = PHt  + 512;         // 512   Kalman gain   [s][m]
    float* KR    = Km   + 512;         // 512   K @ R         [s][m]
    float* Wgj   = KR   + 512;         // 128   8x16 Gauss-Jordan augmented
    float* meanv = Wgj  + 128;         // 64
    float* meanp = meanv + 64;         // 64
    float* resid = meanp + 64;         // 8
    float* fac   = resid + 8;          // 8
    float* piv   = fac   + 8;          // 8 (1 used)

    const int g   = blockIdx.x;
    const int tid = threadIdx.x;

    for (int i = tid; i < 4096; i += 256) {
        Fs[i]   = F[i];
        bufA[i] = P0g[i];              // cov = P0
    }
    for (int i = tid; i < 512; i += 256) Hs[i] = H[i];
    if (tid < 64) {
        Rs[tid]    = Rg[tid];
        meanv[tid] = init_mean[g * S_ + tid];
    }
    __syncthreads();

    float* covP  = bufA;   // covariance at step start
    float* covpP = bufB;   // predicted covariance
    float* tmpP  = bufC;   // matmul intermediate

    for (int t = 0; t < T_; ++t) {
        // ---- predict: tmp = F @ cov ; cov_p = tmp @ F^T + Q ----
        mm64(Fs, covP, tmpP, false, nullptr, nullptr, nullptr);
        __syncthreads();
        mm64(tmpP, Fs, covpP, true, Qg, nullptr, nullptr);
        __syncthreads();

        // ---- PHt = cov_p @ H^T ; mean_p = F @ mean ----
        for (int e = tid; e < 512; e += 256) {
            const int s = e >> 3, m = e & 7;
            float acc = 0.f;
            for (int k = 0; k < 64; ++k) acc += covpP[s * 64 + k] * Hs[m * 64 + k];
            PHt[e] = acc;
        }
        if (tid < 64) {
            float acc = 0.f;
            for (int j = 0; j < 64; ++j) acc += Fs[tid * 64 + j] * meanv[j];
            meanp[tid] = acc;
        }
        __syncthreads();

        // ---- S_inn = H @ PHt + R into augmented [S_inn | I] ; yhat/resid ----
        if (tid < 128) {
            const int i = tid >> 4, c = tid & 15;
            if (c < 8) {
                float acc = Rs[i * 8 + c];
                for (int s = 0; s < 64; ++s) acc += Hs[i * 64 + s] * PHt[s * 8 + c];
                Wgj[i * 16 + c] = acc;
            } else {
                Wgj[i * 16 + c] = (c - 8 == i) ? 1.f : 0.f;
            }
        } else if (tid < 136) {
            const int m = tid - 128;
            float acc = 0.f;
            for (int s = 0; s < 64; ++s) acc += Hs[m * 64 + s] * meanp[s];
            out[(g * T_ + t) * M_ + m] = acc;                 // one-step yhat
            resid[m] = y[(g * T_ + t) * M_ + m] - acc;
        }
        __syncthreads();

        // ---- invert 8x8 S_inn (Gauss-Jordan) ----
        for (int p = 0; p < 8; ++p) {
            if (tid == 0) piv[0] = 1.f / Wgj[p * 16 + p];
            __syncthreads();
            if (tid < 16) Wgj[p * 16 + tid] *= piv[0];
            __syncthreads();
            if (tid < 8) fac[tid] = Wgj[tid * 16 + p];
            __syncthreads();
            if (tid < 128) {
                const int r = tid >> 4, c = tid & 15;
                if (r != p) Wgj[r * 16 + c] -= fac[r] * Wgj[p * 16 + c];
            }
            __syncthreads();
        }

        // ---- K = PHt @ S_inn^{-1} ----
        for (int e = tid; e < 512; e += 256) {
            const int s = e >> 3, m = e & 7;
            float acc = 0.f;
            for (int j = 0; j < 8; ++j) acc += PHt[s * 8 + j] * Wgj[j * 16 + 8 + m];
            Km[e] = acc;
        }
        __syncthreads();

        // ---- mean_u ; KR = K@R ; I_KH into dead cov buffer ----
        if (tid < 64) {
            float acc = meanp[tid];
            for (int m = 0; m < 8; ++m) acc += Km[tid * 8 + m] * resid[m];
            meanv[tid] = acc;
        }
        for (int e = tid; e < 512; e += 256) {
            const int i = e >> 3, m = e & 7;
            float acc = 0.f;
            for (int n = 0; n < 8; ++n) acc += Km[i * 8 + n] * Rs[n * 8 + m];
            KR[e] = acc;
        }
        for (int e = tid; e < 4096; e += 256) {
            const int i = e >> 6, j = e & 63;
            float acc = (i == j) ? 1.f : 0.f;
            for (int m = 0; m < 8; ++m) acc -= Km[i * 8 + m] * Hs[m * 64 + j];
            covP[e] = acc;                      // covP now holds I - K H
        }
        __syncthreads();

        // ---- Joseph form: cov = IKH @ cov_p @ IKH^T + K R K^T ----
        mm64(covP, covpP, tmpP, false, nullptr, nullptr, nullptr);
        __syncthreads();
        mm64(tmpP, covP, covpP, true, nullptr, KR, Km);   // fused + KR*K^T
        __syncthreads();

        float* sw = covP; covP = covpP; covpP = sw;       // rotate buffers
    }
}

extern "C" void kernel_launch(void* const* d_in, const int* in_sizes, int n_in,
                              void* d_out, int out_size, void* d_ws, size_t ws_size,
                              hipStream_t stream) {
    const float* y         = (const float*)d_in[0];
    const float* F         = (const float*)d_in[1];
    const float* H         = (const float*)d_in[2];
    const float* Lq        = (const float*)d_in[3];
    const float* Lr        = (const float*)d_in[4];
    const float* init_mean = (const float*)d_in[5];
    const float* Linit     = (const float*)d_in[6];
    float* out = (float*)d_out;
    float* ws  = (float*)d_ws;

    kf_prep<<<16, 256, 0, stream>>>(Lq, Lr, Linit, ws);

    // dynamic LDS: F + 3x(64x64) + small scratch
    const size_t smem_bytes = (4096 * 4 + 512 + 64 + 512 + 512 + 512 + 128 +
                               64 + 64 + 8 + 8 + 8) * sizeof(float);
    kf_main<<<G_, 256, smem_bytes, stream>>>(y, F, H,
                                             ws,            // Q
                                             ws + 8192,     // R
                                             ws + 4096,     // P0
                                             init_mean, out);
}